// QuantumSentenceTransformer_13752485282220
// MI455X (gfx1250) — compile-verified
//
#include <hip/hip_runtime.h>
#include <math.h>

// QuantumSentenceTransformer fused kernel for MI455X (gfx1250, wave32).
//
// Latency-bound problem (<10KB I/O, ~5 MFLOP total): one fused kernel, one
// wave per batch element (32 blocks x 32 threads), zero LDS / global
// intermediates. 512 real amplitudes live entirely in 16 VGPRs/lane:
// idx = r*32 + lane. Wire w acts on bit p = 8-w:
//   wires 0..3 -> register bits (per-lane FMA / free register renaming)
//   wires 4..8 -> lane bits (__shfl_xor -> ds_swizzle/v_permlane)
// Pre-layer GEMM (32x54 @ 54x9) done per-block with V_WMMA_F32_16X16X4_F32
// (14 WMMAs, fp32, K padded 54->56, N padded 9->16). OOB guards implemented
// as clamped-address loads + v_cndmask zeroing -> no exec-mask branching.
// Gate angles are tiny (|a| <= pi/4) so sin/cos use the native
// V_SIN_F32/V_COS_F32 path (__sinf/__cosf) instead of libm range reduction.

typedef __attribute__((ext_vector_type(2))) float v2f;
typedef __attribute__((ext_vector_type(8))) float v8f;

#define NQ     9
#define QDEPTH 6
#define BATCH  32
#define KDIM   54   // 6*9 input features

__device__ __forceinline__ float shflx(float v, int mask) {
    return __shfl_xor(v, mask, 32);
}

__device__ __forceinline__ void swapf(float& a, float& b) {
    float t = a; a = b; b = t;
}

// Unconditional load with clamped address, then select-to-zero (no branches).
__device__ __forceinline__ float guarded_load(const float* __restrict__ p,
                                              int idx, bool ok) {
    const float v = p[ok ? idx : 0];   // address cndmask, load with full EXEC
    return ok ? v : 0.f;               // value cndmask
}

// RY on a register-bit wire: pairs (r, r|1<<RB), pure per-lane VALU.
template <int RB>
__device__ __forceinline__ void ry_regbit(float* s, float c, float sn) {
#pragma unroll
    for (int r = 0; r < 16; ++r) {
        if ((r & (1 << RB)) == 0) {
            const int r1 = r | (1 << RB);
            const float a0 = s[r], a1 = s[r1];
            s[r]  = c * a0 - sn * a1;
            s[r1] = sn * a0 + c * a1;
        }
    }
}

// RY on a lane-bit wire: partner via shfl_xor; sign picked by own lane bit.
template <int LB>
__device__ __forceinline__ void ry_lanebit(float* s, float c, float sn, int lane) {
    const float sgn = (lane & (1 << LB)) ? sn : -sn;
#pragma unroll
    for (int r = 0; r < 16; ++r) {
        const float p = shflx(s[r], 1 << LB);
        s[r] = c * s[r] + sgn * p;
    }
}

// One full RY layer over wires 0..8 with per-wire angles th[].
// Half-angles are bounded well inside [-pi/4, pi/4], so native
// V_SIN_F32/V_COS_F32 (__sinf/__cosf) are ulp-accurate here.
__device__ __forceinline__ void ry_all(float* s, const float* th, int lane) {
    { const float a = 0.5f * th[0]; ry_regbit<3>(s, __cosf(a), __sinf(a)); }        // wire 0, bit 8
    { const float a = 0.5f * th[1]; ry_regbit<2>(s, __cosf(a), __sinf(a)); }        // wire 1, bit 7
    { const float a = 0.5f * th[2]; ry_regbit<1>(s, __cosf(a), __sinf(a)); }        // wire 2, bit 6
    { const float a = 0.5f * th[3]; ry_regbit<0>(s, __cosf(a), __sinf(a)); }        // wire 3, bit 5
    { const float a = 0.5f * th[4]; ry_lanebit<4>(s, __cosf(a), __sinf(a), lane); } // wire 4, bit 4
    { const float a = 0.5f * th[5]; ry_lanebit<3>(s, __cosf(a), __sinf(a), lane); } // wire 5, bit 3
    { const float a = 0.5f * th[6]; ry_lanebit<2>(s, __cosf(a), __sinf(a), lane); } // wire 6, bit 2
    { const float a = 0.5f * th[7]; ry_lanebit<1>(s, __cosf(a), __sinf(a), lane); } // wire 7, bit 1
    { const float a = 0.5f * th[8]; ry_lanebit<0>(s, __cosf(a), __sinf(a), lane); } // wire 8, bit 0
}

// CNOT where control lane-bit CB selects a lane-swap across target bit TB.
template <int CB, int TB>
__device__ __forceinline__ void cnot_lane(float* s, int lane) {
    const bool ctl = (lane & (1 << CB)) != 0;
#pragma unroll
    for (int r = 0; r < 16; ++r) {
        const float p = shflx(s[r], 1 << TB);
        s[r] = ctl ? p : s[r];
    }
}

// Both CNOT sublayers of one depth step.
__device__ __forceinline__ void cnot_layers(float* s, int lane) {
    // ---- even pairs: (0,1) (2,3) (4,5) (6,7) ----
    // CNOT(0,1): ctrl reg bit3, tgt reg bit2 -> swap r <-> r^4 where bit3 set
    swapf(s[8], s[12]); swapf(s[9], s[13]); swapf(s[10], s[14]); swapf(s[11], s[15]);
    // CNOT(2,3): ctrl reg bit1, tgt reg bit0 -> swap r <-> r^1 where bit1 set
    swapf(s[2], s[3]); swapf(s[6], s[7]); swapf(s[10], s[11]); swapf(s[14], s[15]);
    // CNOT(4,5): ctrl lane bit4, tgt lane bit3
    cnot_lane<4, 3>(s, lane);
    // CNOT(6,7): ctrl lane bit2, tgt lane bit1
    cnot_lane<2, 1>(s, lane);
    // ---- odd pairs: (1,2) (3,4) (5,6) (7,8) ----
    // CNOT(1,2): ctrl reg bit2, tgt reg bit1 -> swap r <-> r^2 where bit2 set
    swapf(s[4], s[6]); swapf(s[5], s[7]); swapf(s[12], s[14]); swapf(s[13], s[15]);
    // CNOT(3,4): ctrl reg bit0 (r odd), tgt lane bit4 -> unconditional lane swap for odd r
#pragma unroll
    for (int r = 1; r < 16; r += 2) s[r] = shflx(s[r], 16);
    // CNOT(5,6): ctrl lane bit3, tgt lane bit2
    cnot_lane<3, 2>(s, lane);
    // CNOT(7,8): ctrl lane bit1, tgt lane bit0
    cnot_lane<1, 0>(s, lane);
}

__global__ __launch_bounds__(32)
void qst_kernel(const float* __restrict__ x,       // [32, 54]
                const float* __restrict__ pre_w,   // [9, 54]
                const float* __restrict__ pre_b,   // [9]
                const float* __restrict__ qp,      // [54] = [6, 9]
                const float* __restrict__ post_w,  // [9, 9]
                const float* __restrict__ post_b,  // [9]
                float* __restrict__ out)           // [32, 9]
{
    const int b    = blockIdx.x;        // batch element
    const int lane = threadIdx.x;       // 0..31 (wave32)

    // ---------------- pre-layer: x @ pre_w.T via V_WMMA_F32_16X16X4_F32 ----
    // This block computes the 16-row tile containing its batch row.
    const int t    = b >> 4;                    // tile 0: rows 0-15, tile 1: rows 16-31
    const int m    = 16 * t + (lane & 15);      // A-matrix M per ISA layout
    const int n    = lane & 15;                 // B/C N per ISA layout
    const int ksel = (lane < 16) ? 0 : 2;       // lanes 16-31 carry K+2,K+3
    const bool nok = (n < NQ);                  // pad N 9 -> 16 with zeros

    v8f acc = {0.f, 0.f, 0.f, 0.f, 0.f, 0.f, 0.f, 0.f};
#pragma unroll
    for (int k0 = 0; k0 < KDIM; k0 += 4) {      // 14 iterations, K padded to 56
        const int ka = k0 + ksel;
        const bool k0ok = (ka     < KDIM);
        const bool k1ok = (ka + 1 < KDIM);
        v2f av, bv;
        av.x = guarded_load(x, m * KDIM + ka,     k0ok);
        av.y = guarded_load(x, m * KDIM + ka + 1, k1ok);
        bv.x = guarded_load(pre_w, n * KDIM + ka,     nok && k0ok);
        bv.y = guarded_load(pre_w, n * KDIM + ka + 1, nok && k1ok);
        acc = __builtin_amdgcn_wmma_f32_16x16x4_f32(
            /*neg_a=*/false, av, /*neg_b=*/false, bv,
            /*c_mod=*/(short)0, acc, /*reuse_a=*/false, /*reuse_b=*/false);
    }

    // Extract our batch row from the C tile (VGPR v: lanes0-15 row v, lanes16-31 row v+8)
    const int r      = b & 15;
    const int rowsel = r & 7;
    const int half16 = (r >> 3) ? 16 : 0;
    float rowv = acc[0];
#pragma unroll
    for (int v = 1; v < 8; ++v) rowv = (rowsel == v) ? acc[v] : rowv;

    float th[NQ];
#pragma unroll
    for (int w = 0; w < NQ; ++w) {
        const float pv = __shfl(rowv, w + half16, 32) + pre_b[w];
        th[w] = tanhf(pv) * 1.5707963267948966f;     // pi/2
    }

    // ---------------- state-vector simulation, fully register-resident ------
    float s[16];
#pragma unroll
    for (int rr = 0; rr < 16; ++rr) s[rr] = 0.04419417382415922f;  // 1/sqrt(512)

    ry_all(s, th, lane);                              // input-angle RY layer

    for (int k = 0; k < QDEPTH; ++k) {
        cnot_layers(s, lane);
        float thw[NQ];
#pragma unroll
        for (int w = 0; w < NQ; ++w) thw[w] = qp[k * NQ + w];  // wave-uniform (SMEM)
        ry_all(s, thw, lane);
    }

    // ---------------- <Z_w> = sum probs * (1 - 2*bit_{8-w}) ----------------
    float p2[16];
    float tot = 0.f;
#pragma unroll
    for (int rr = 0; rr < 16; ++rr) { p2[rr] = s[rr] * s[rr]; tot += p2[rr]; }

    float z[NQ];
#pragma unroll
    for (int w = 0; w < 4; ++w) {                     // sign from register bit (3-w)
        const int rb = 3 - w;
        float a = 0.f;
#pragma unroll
        for (int rr = 0; rr < 16; ++rr) a += ((rr >> rb) & 1) ? -p2[rr] : p2[rr];
#pragma unroll
        for (int off = 16; off > 0; off >>= 1) a += shflx(a, off);
        z[w] = a;
    }
#pragma unroll
    for (int w = 4; w < NQ; ++w) {                    // sign from lane bit (8-w)
        const int lb = 8 - w;
        float a = ((lane >> lb) & 1) ? -tot : tot;
#pragma unroll
        for (int off = 16; off > 0; off >>= 1) a += shflx(a, off);
        z[w] = a;
    }

    // ---------------- post-layer: z @ post_w.T + post_b (lane j = output j) --
    if (lane < NQ) {
        float o = post_b[lane];
#pragma unroll
        for (int w = 0; w < NQ; ++w) o += z[w] * post_w[lane * NQ + w];
        out[b * NQ + lane] = o;
    }
}

extern "C" void kernel_launch(void* const* d_in, const int* in_sizes, int n_in,
                              void* d_out, int out_size, void* d_ws, size_t ws_size,
                              hipStream_t stream) {
    (void)in_sizes; (void)n_in; (void)out_size; (void)d_ws; (void)ws_size;
    const float* x      = (const float*)d_in[0];   // [32, 54]
    const float* pre_w  = (const float*)d_in[1];   // [9, 54]
    const float* pre_b  = (const float*)d_in[2];   // [9]
    const float* qp     = (const float*)d_in[3];   // [54]
    const float* post_w = (const float*)d_in[4];   // [9, 9]
    const float* post_b = (const float*)d_in[5];   // [9]
    qst_kernel<<<dim3(BATCH), dim3(32), 0, stream>>>(
        x, pre_w, pre_b, qp, post_w, post_b, (float*)d_out);
}